// SO3ToS2_4784593568435
// MI455X (gfx1250) — compile-verified
//
#include <hip/hip_runtime.h>
#include <math.h>

// ---------------------------------------------------------------------------
// SO(3) -> S2 spherical convolution for MI455X (gfx1250), wave32 + WMMA f32
// + async global->LDS double buffering.
//
//   T0 twiddle table (transposed [freq 32][spatial 64] cos/sin)
//   T1 wig[t][tri]  = d^l_{m,k}(beta_in[t]) * quad_w[t]   (tri packs l,m,k)
//   T1b triIdx[tri] = packed (m_idx<<5 | k_idx) for LDS gather
//   T2 d_inv[l][t][m] = (2l+1) d^l_{m,0}(beta_out[t])
//   T3 yDsum[l][k][kg] = e^{i m_k g} * sum_n e^{i m_n a} d^l_{n,k}(b)
//   T4 Q[2*lm+c][t*32+p] = d_inv * {cos,-sin}(2 pi m p/32)
//   K2 ysum = SCALING * kernel . yDsum
//   K1 fused DFT(gamma)->DFT(alpha)->beta quadrature, 1 WG per (B,Fi),
//      x read once via GLOBAL_LOAD_ASYNC_TO_LDS_B128 double buffering.
//   K3 zs = conj(xhat) . ysum (complex WMMA)
//   K4 out = zs[256x512] x Q[512x1024] + bias (WMMA)
// ---------------------------------------------------------------------------

typedef float v2f __attribute__((ext_vector_type(2)));
typedef float v8f __attribute__((ext_vector_type(8)));

#define WMMA_F32(a, b, c) \
  __builtin_amdgcn_wmma_f32_16x16x4_f32(false, (a), false, (b), (short)0, (c), false, false)

#define PI_D 3.14159265358979323846
#define SCALING_F 0.17677669529663688f   // 1/sqrt(16 * 16*16^3/32^3) = 1/sqrt(32)

#if defined(__gfx1250__) && __has_builtin(__builtin_amdgcn_global_load_async_to_lds_b128)
#define HAVE_ASYNC_LDS 1
// exact param types per hipcc diagnostic: int __vector(4) ptrs in AS1 / AS3
typedef int vint4 __attribute__((vector_size(16)));
typedef __attribute__((address_space(1))) vint4 gvint4;
typedef __attribute__((address_space(3))) vint4 svint4;
#else
#define HAVE_ASYNC_LDS 0
#endif

#if defined(__gfx1250__) && __has_builtin(__builtin_amdgcn_s_wait_asynccnt)
#define WAIT_ASYNC(n) __builtin_amdgcn_s_wait_asynccnt(n)
#else
#define WAIT_ASYNC(n) asm volatile("s_wait_asynccnt %0" ::"i"(n) : "memory")
#endif

__constant__ int c_S2[17] = {0,1,10,35,84,165,286,455,680,969,1330,1771,2300,2925,3654,4495,5456};

// workspace layout (float offsets)
constexpr size_t OFF_COSW = 0;              // 2048  ([32 freq][64 spatial])
constexpr size_t OFF_SINW = 2048;           // 2048
constexpr size_t OFF_WIG  = 4096;           // 349184 = 64*5456, layout [t][tri]
constexpr size_t OFF_DINV = 353280;         // 8192
constexpr size_t OFF_YDS  = 361472;         // 8192  (4096 float2)
constexpr size_t OFF_Q    = 369664;         // 524288
constexpr size_t OFF_YSUM = 893952;         // 262144 (131072 float2)
constexpr size_t OFF_XHAT = 1156096;        // 1396736 (698368 float2)
constexpr size_t OFF_A    = 2552832;        // 131072
constexpr size_t OFF_TRI  = 2683904;        // 5456 uint (pad to 5504)
constexpr size_t WS_FLOATS = 2689408;       // ~10.8 MB

// -------------------------------- helpers ----------------------------------

__device__ __forceinline__ int isqrt_i(int v) {
  int l = (int)sqrtf((float)v);
  while ((l + 1) * (l + 1) <= v) ++l;
  while (l * l > v) --l;
  return l;
}

__device__ double wig_d(int l, int m, int n, double beta) {
  double f[32];
  f[0] = 1.0;
  for (int i = 1; i < 32; ++i) f[i] = f[i - 1] * (double)i;
  double cb = cos(0.5 * beta), sb = sin(0.5 * beta);
  double pref = sqrt(f[l + m] * f[l - m] * f[l + n] * f[l - n]);
  int s0 = (n - m) > 0 ? (n - m) : 0;
  int s1 = (l + n) < (l - m) ? (l + n) : (l - m);
  double acc = 0.0;
  for (int s = s0; s <= s1; ++s) {
    double t = pow(cb, (double)(2 * l + n - m - 2 * s)) * pow(sb, (double)(m - n + 2 * s)) /
               (f[l + n - s] * f[s] * f[m - n + s] * f[l - m - s]);
    acc += ((m - n + s) & 1) ? -t : t;
  }
  return pref * acc;
}

__device__ double quad_w(int j) {  // SOFT weights, b = 32, folded /(2b)^2
  double s = 0.0;
  for (int k = 0; k < 32; ++k)
    s += sin((double)(2 * j + 1) * (double)(2 * k + 1) * PI_D / 128.0) / (double)(2 * k + 1);
  return (2.0 / 32.0) * sin(PI_D * (double)(2 * j + 1) / 128.0) * s / 4096.0;
}

// ------------------------------ table builders -----------------------------

// transposed twiddle: cw[n*64+g] = cos(2 pi (n-15) g / 64), n<32, g<64
__global__ void k_twiddle(float* __restrict__ cw, float* __restrict__ sw) {
  int idx = blockIdx.x * blockDim.x + threadIdx.x;
  if (idx >= 2048) return;
  int n = idx >> 6, g = idx & 63;
  double ang = 2.0 * PI_D * (double)(n - 15) * (double)g / 64.0;
  cw[idx] = (float)cos(ang);
  sw[idx] = (float)sin(ang);
}

__global__ void k_triidx(unsigned* __restrict__ tri) {
  int t = blockIdx.x * blockDim.x + threadIdx.x;
  if (t >= 5456) return;
  int l = 0;
  for (int q = 1; q < 16; ++q)
    if (t >= c_S2[q]) l = q;
  int rem = t - c_S2[l], tl = 2 * l + 1;
  int m_local = rem / tl, k_local = rem % tl;
  unsigned m_idx = (unsigned)(m_local + 15 - l), k_idx = (unsigned)(k_local + 15 - l);
  tri[t] = (m_idx << 5) | k_idx;
}

__global__ void k_wig(float* __restrict__ wig) {  // layout [t][tri]
  int idx = blockIdx.x * blockDim.x + threadIdx.x;
  if (idx >= 349184) return;
  int t = idx / 5456, tri = idx - t * 5456;
  int l = 0;
  for (int q = 1; q < 16; ++q)
    if (tri >= c_S2[q]) l = q;
  int rem = tri - c_S2[l], tl = 2 * l + 1;
  int m_local = rem / tl, k_local = rem % tl;
  double beta = PI_D * (double)(2 * t + 1) / 128.0;
  wig[idx] = (float)(wig_d(l, m_local - l, k_local - l, beta) * quad_w(t));
}

__global__ void k_dinv(float* __restrict__ dinv) {
  int idx = blockIdx.x * blockDim.x + threadIdx.x;
  if (idx >= 8192) return;
  int l = isqrt_i(idx >> 5);
  int rem = idx - 32 * l * l;
  int tl = 2 * l + 1;
  int t = rem / tl, m_local = rem % tl;
  double beta = PI_D * (double)(2 * t + 1) / 64.0;
  dinv[idx] = (float)((double)tl * wig_d(l, m_local - l, 0, beta));
}

__global__ void k_yds(float2* __restrict__ yds) {
  int idx = blockIdx.x * blockDim.x + threadIdx.x;
  if (idx >= 4096) return;
  int rowk = idx >> 4, kg = idx & 15;
  int l = isqrt_i(rowk);
  int k_local = rowk - l * l;
  int b_idx = kg >> 3, a_idx = (kg >> 1) & 3, g_idx = kg & 1;
  double beta = b_idx ? PI_D / 8.0 : PI_D / 16.0;
  double alpha = (double)a_idx * PI_D * 0.5;
  double gamma = (double)g_idx * PI_D;
  int mk = k_local - l;
  double sr = 0.0, si = 0.0;
  for (int n = 0; n < 2 * l + 1; ++n) {
    int mn = n - l;
    double d = wig_d(l, mn, mk, beta);
    double ph = (double)mn * alpha;           // conj(D): e^{+i mn alpha}
    sr += d * cos(ph);
    si += d * sin(ph);
  }
  double cg = cos((double)mk * gamma), sg = sin((double)mk * gamma);
  yds[idx] = make_float2((float)(sr * cg - si * sg), (float)(sr * sg + si * cg));
}

__global__ void k_Q(const float* __restrict__ dinv, float* __restrict__ Q) {
  int idx = blockIdx.x * blockDim.x + threadIdx.x;
  if (idx >= 524288) return;
  int rowq = idx >> 10, col = idx & 1023;
  int lm = rowq >> 1, comp = rowq & 1;
  int l = isqrt_i(lm);
  int m_local = lm - l * l;
  int t = col >> 5, p = col & 31;
  float d = dinv[32 * l * l + t * (2 * l + 1) + m_local];
  float th = (float)(2.0 * PI_D) * (float)(m_local - l) * (float)p / 32.0f;
  Q[idx] = comp ? (-d * sinf(th)) : (d * cosf(th));
}

// K2: ysum[l][k][i][o] = SCALING * sum_kg kernel[i,o,kg] * yDsum[l][k][kg]
__global__ void k_ysum(const float* __restrict__ ker, const float2* __restrict__ yds,
                       float2* __restrict__ ysum) {
  int e = blockIdx.x * blockDim.x + threadIdx.x;
  if (e >= 131072) return;
  int rowk = e >> 9;            // l^2 + k_local
  int io = e & 511;
  int i = io >> 5, o = io & 31;
  float sr = 0.f, si = 0.f;
  for (int kg = 0; kg < 16; ++kg) {
    float kv = ker[(i * 32 + o) * 16 + kg];
    float2 y = yds[rowk * 16 + kg];
    sr += kv * y.x;
    si += kv * y.y;
  }
  ysum[e] = make_float2(SCALING_F * sr, SCALING_F * si);
}

// --------- K1: fused DFT(gamma) -> DFT(alpha) -> beta quadrature -----------
// One workgroup (8 waves) per (B, Fi). x read exactly once from HBM via
// double-buffered async global->LDS B128 DMA.

__global__ __launch_bounds__(256) void k_so3fft_xhat(const float* __restrict__ x,
                                                     const float* __restrict__ cw,
                                                     const float* __restrict__ sw,
                                                     const float* __restrict__ wig,
                                                     const unsigned* __restrict__ triIdx,
                                                     float2* __restrict__ xhat) {
  __shared__ float xs[2][64][68];   // stride 68: 16B-aligned rows, conflict-free
  __shared__ float tRe[64][33];     // stage-1 result (gamma DFT)
  __shared__ float tIm[64][33];
  __shared__ float xcRe[32][33];    // stage-2 result (alpha DFT)
  __shared__ float xcIm[32][33];
  __shared__ float xhRe[5456];      // per-(B,Fi) SO3 spectrum accumulators
  __shared__ float xhIm[5456];

  const int tid = threadIdx.x;
  const int wave = tid >> 5, lane = tid & 31;
  const int hf = lane >> 4, r = lane & 15;
  const int bi = blockIdx.x;        // B*16 + Fi

  for (int i = tid; i < 5456; i += 256) { xhRe[i] = 0.f; xhIm[i] = 0.f; }

  const float* xbase = x + (size_t)bi * 64 * 4096;
  const float2* cw2 = (const float2*)cw;
  const float2* sw2 = (const float2*)sw;

#if HAVE_ASYNC_LDS
  // prologue: async-DMA slab 0 into buffer 0 (4 x B128 per wave)
  for (int j = 0; j < 4; ++j) {
    int i = tid + j * 256;                       // float4 index
    int a = i >> 4, g4 = (i & 15) << 2;
    __builtin_amdgcn_global_load_async_to_lds_b128(
        (gvint4*)(xbase + (size_t)i * 4),
        (svint4*)&xs[0][a][g4], 0, 0);
  }
#endif
  __syncthreads();

  for (int t = 0; t < 64; ++t) {
    const int buf = t & 1;
#if HAVE_ASYNC_LDS
    if (t + 1 < 64) {                            // DMA next slab while computing
      for (int j = 0; j < 4; ++j) {
        int i = tid + j * 256;
        int a = i >> 4, g4 = (i & 15) << 2;
        __builtin_amdgcn_global_load_async_to_lds_b128(
            (gvint4*)(xbase + (size_t)(t + 1) * 4096 + (size_t)i * 4),
            (svint4*)&xs[buf ^ 1][a][g4], 0, 0);
      }
      WAIT_ASYNC(4);                             // oldest 4 (slab t) complete
    } else {
      WAIT_ASYNC(0);
    }
#else
    for (int i = tid; i < 1024; i += 256) {
      float4 v = ((const float4*)(xbase + (size_t)t * 4096))[i];
      int a = i >> 4, g4 = (i & 15) << 2;
      xs[buf][a][g4 + 0] = v.x; xs[buf][a][g4 + 1] = v.y;
      xs[buf][a][g4 + 2] = v.z; xs[buf][a][g4 + 3] = v.w;
    }
    if (t + 1 < 64) {
      const char* p = (const char*)(xbase + (size_t)(t + 1) * 4096);
      __builtin_prefetch(p + tid * 64, 0, 1);
    }
#endif
    __syncthreads();

    // stage 1: tmp[a,n] = x[a,:] . e^{-i th(n,g)}   (2 real matmuls 64x32x64)
    for (int tj = wave; tj < 16; tj += 8) {
      int part = tj >> 3, tile = tj & 7, mt = tile >> 1, nt = tile & 1;
      const float2* W2 = part ? sw2 : cw2;
      v8f c = {};
      for (int k0 = 0; k0 < 64; k0 += 4) {
        int ka = k0 + 2 * hf;
        v2f a, b;
        a[0] = xs[buf][mt * 16 + r][ka];
        a[1] = xs[buf][mt * 16 + r][ka + 1];
        float2 wv = W2[((nt * 16 + r) * 64 + ka) >> 1];   // B[K=g][N=n] pair
        b[0] = wv.x; b[1] = wv.y;
        c = WMMA_F32(a, b, c);
      }
      for (int j = 0; j < 8; ++j) {
        int row = mt * 16 + j + 8 * hf, col = nt * 16 + r;
        if (part) tIm[row][col] = -c[j]; else tRe[row][col] = c[j];
      }
    }
    __syncthreads();

    // stage 2: Xc[m,n] = E(m,:) . tmp   (complex; 4 real matmuls 32x32x64)
    {
      int part = wave >> 2;            // 0 = Re, 1 = Im
      int tile = wave & 3, mt = tile >> 1, nt = tile & 1;
      v8f c = {};
      for (int k0 = 0; k0 < 64; k0 += 4) {     // pass 1: cosE . (Re|Im)
        int ka = k0 + 2 * hf;
        v2f a, b;
        float2 av = cw2[((mt * 16 + r) * 64 + ka) >> 1];  // A[M=m][K=a] pair
        a[0] = av.x; a[1] = av.y;
        const float(*S)[33] = part ? tIm : tRe;
        b[0] = S[ka][nt * 16 + r];
        b[1] = S[ka + 1][nt * 16 + r];
        c = WMMA_F32(a, b, c);
      }
      for (int k0 = 0; k0 < 64; k0 += 4) {     // pass 2: (+sin.Im) | (-sin.Re)
        int ka = k0 + 2 * hf;
        v2f a, b;
        float2 av = sw2[((mt * 16 + r) * 64 + ka) >> 1];
        float sgn = part ? -1.f : 1.f;
        a[0] = sgn * av.x; a[1] = sgn * av.y;
        const float(*S)[33] = part ? tRe : tIm;
        b[0] = S[ka][nt * 16 + r];
        b[1] = S[ka + 1][nt * 16 + r];
        c = WMMA_F32(a, b, c);
      }
      for (int j = 0; j < 8; ++j) {
        int row = mt * 16 + j + 8 * hf, col = nt * 16 + r;
        if (part) xcIm[row][col] = c[j]; else xcRe[row][col] = c[j];
      }
    }
    __syncthreads();

    // stage 3: beta quadrature accumulate (table-driven, coalesced wig reads)
    for (int tri = tid; tri < 5456; tri += 256) {
      unsigned mi = triIdx[tri];
      float w = wig[t * 5456 + tri];
      xhRe[tri] += w * xcRe[mi >> 5][mi & 31];
      xhIm[tri] += w * xcIm[mi >> 5][mi & 31];
    }
    __syncthreads();
  }

  for (int tri = tid; tri < 5456; tri += 256)
    xhat[(size_t)tri * 128 + bi] = make_float2(xhRe[tri], xhIm[tri]);
}

// ------ K3: zs[(m,B),(o)] = sum_{k,i} conj(xhat) * ysum  (complex WMMA) ----

__global__ __launch_bounds__(256) void k_zs(const float2* __restrict__ xhat,
                                            const float2* __restrict__ ysum,
                                            float* __restrict__ Aout) {
  const int l = blockIdx.x;
  const int tl = 2 * l + 1;
  const int R = tl * 8, KK = tl * 16;
  const int rtiles = (R + 15) >> 4;
  const int wave = threadIdx.x >> 5, lane = threadIdx.x & 31;
  const int hf = lane >> 4, r = lane & 15;
  const int s2l = c_S2[l], l2 = l * l;

  for (int job = wave; job < rtiles * 2; job += 8) {
    int rt = job >> 1, ct = job & 1;
    v8f cRe = {}, cIm = {}, cIm2 = {};
    int rowA = rt * 16 + r;
    if (rowA >= R) rowA = R - 1;               // clamp (guarded on store)
    int m_loc = rowA >> 3, Bb = rowA & 7;
    for (int kk0 = 0; kk0 < KK; kk0 += 4) {
      v2f xr, xi, yr, yi;
#pragma unroll
      for (int ii = 0; ii < 2; ++ii) {
        int kk = kk0 + 2 * hf + ii;
        int k_loc = kk >> 4, ci = kk & 15;
        float2 xv = xhat[(size_t)(s2l + m_loc * tl + k_loc) * 128 + Bb * 16 + ci];
        xr[ii] = xv.x; xi[ii] = xv.y;
        float2 yv = ysum[(size_t)(l2 + k_loc) * 512 + ci * 32 + ct * 16 + r];
        yr[ii] = yv.x; yi[ii] = yv.y;
      }
      cRe  = WMMA_F32(xr, yr, cRe);   // conj(x)*y: Re = xr*yr + xi*yi
      cRe  = WMMA_F32(xi, yi, cRe);
      cIm  = WMMA_F32(xr, yi, cIm);   // Im = xr*yi - xi*yr
      cIm2 = WMMA_F32(xi, yr, cIm2);
    }
    for (int j = 0; j < 8; ++j) {
      int row = rt * 16 + j + 8 * hf;
      if (row < R) {
        int ml = row >> 3, Bb2 = row & 7, o = ct * 16 + r;
        int lm = l2 + ml;
        Aout[(size_t)(Bb2 * 32 + o) * 512 + 2 * lm + 0] = cRe[j];
        Aout[(size_t)(Bb2 * 32 + o) * 512 + 2 * lm + 1] = cIm[j] - cIm2[j];
      }
    }
  }
}

// -------- K4: out[256,1024] = A[256,512] x Q[512,1024] + bias (WMMA) -------

__global__ __launch_bounds__(256) void k_out(const float* __restrict__ A,
                                             const float* __restrict__ Q,
                                             const float* __restrict__ bias,
                                             float* __restrict__ out) {
  const int wave = threadIdx.x >> 5, lane = threadIdx.x & 31;
  const int hf = lane >> 4, r = lane & 15;
  int tile = blockIdx.x * 8 + wave;       // 1024 tiles: 16 row-tiles x 64 col-tiles
  int rt = tile >> 6, ct = tile & 63;
  const float2* A2 = (const float2*)A;
  v8f c = {};
  for (int k0 = 0; k0 < 512; k0 += 4) {
    int ka = k0 + 2 * hf;
    v2f a, b;
    float2 av = A2[((size_t)(rt * 16 + r) * 512 + ka) >> 1];
    a[0] = av.x; a[1] = av.y;
    b[0] = Q[(size_t)(ka + 0) * 1024 + ct * 16 + r];
    b[1] = Q[(size_t)(ka + 1) * 1024 + ct * 16 + r];
    c = WMMA_F32(a, b, c);
  }
  for (int j = 0; j < 8; ++j) {
    int row = rt * 16 + j + 8 * hf, col = ct * 16 + r;
    out[(size_t)row * 1024 + col] = c[j] + bias[row & 31];
  }
}

// ------------------------------- launcher ----------------------------------

extern "C" void kernel_launch(void* const* d_in, const int* in_sizes, int n_in,
                              void* d_out, int out_size, void* d_ws, size_t ws_size,
                              hipStream_t stream) {
  (void)in_sizes; (void)n_in; (void)out_size;
  if (ws_size < WS_FLOATS * sizeof(float)) return;

  const float* x    = (const float*)d_in[0];
  const float* ker  = (const float*)d_in[1];
  const float* bias = (const float*)d_in[2];
  float* ws = (float*)d_ws;

  float*    cw   = ws + OFF_COSW;
  float*    sw   = ws + OFF_SINW;
  float*    wig  = ws + OFF_WIG;
  float*    dinv = ws + OFF_DINV;
  float2*   yds  = (float2*)(ws + OFF_YDS);
  float*    Q    = ws + OFF_Q;
  float2*   ysum = (float2*)(ws + OFF_YSUM);
  float2*   xhat = (float2*)(ws + OFF_XHAT);
  float*    A    = ws + OFF_A;
  unsigned* tri  = (unsigned*)(ws + OFF_TRI);

  k_twiddle<<<8, 256, 0, stream>>>(cw, sw);
  k_triidx<<<22, 256, 0, stream>>>(tri);
  k_wig<<<(349184 + 255) / 256, 256, 0, stream>>>(wig);
  k_dinv<<<32, 256, 0, stream>>>(dinv);
  k_yds<<<16, 256, 0, stream>>>(yds);
  k_Q<<<2048, 256, 0, stream>>>(dinv, Q);
  k_ysum<<<512, 256, 0, stream>>>(ker, yds, ysum);
  k_so3fft_xhat<<<128, 256, 0, stream>>>(x, cw, sw, wig, tri, xhat);
  k_zs<<<16, 256, 0, stream>>>(xhat, ysum, A);
  k_out<<<128, 256, 0, stream>>>(A, Q, bias, (float*)d_out);
}